// Cross_Attention_Local_31834297598519
// MI455X (gfx1250) — compile-verified
//
#include <hip/hip_runtime.h>

// CDNA5 (gfx1250) fused channel cross-attention, fp32 WMMA 16x16x4.
// One workgroup (256 threads = 8 wave32) per patch b; loop over 8 heads.
// All intermediates in LDS/registers; weights stream from L2 (2MB hot set).

typedef __attribute__((ext_vector_type(2))) float v2f;
typedef __attribute__((ext_vector_type(8))) float v8f;

__device__ __forceinline__ v8f wmma4(v2f a, v2f b, v8f c) {
  // V_WMMA_F32_16X16X4_F32: D = A(16x4) * B(4x16) + C(16x16)
  return __builtin_amdgcn_wmma_f32_16x16x4_f32(
      /*neg_a=*/false, a, /*neg_b=*/false, b,
      /*c_mod=*/(short)0, c, /*reuse_a=*/false, /*reuse_b=*/false);
}

#define IN_EPS 1e-5f

__global__ __launch_bounds__(256) void fused_cross_attn_local(
    const float* __restrict__ emb, const float* __restrict__ Wq,
    const float* __restrict__ Wk, const float* __restrict__ Wv,
    const float* __restrict__ Wout, float* __restrict__ out) {
  // ---- static LDS: 56 KB total ----
  __shared__ float embA[8 * 128];    // emb[b] row-major  [p][j]
  __shared__ float embT[128 * 8];    // emb[b]^T          [j][p]
  __shared__ float qT[128 * 8];      // q_bh^T            [c][p]
  __shared__ float kB[8 * 128];      // k_bh              [p][d]
  __shared__ float vT[128 * 8];      // v_bh^T            [d][p]
  __shared__ float ctxT[8 * 128];    // ctx^T             [p][c]
  __shared__ float attnL[128 * 64];  // attn K-half stage [c][k]
  __shared__ float redbuf[16];       // cross-wave sum/sumsq

  const int b = blockIdx.x;
  const int tid = (int)threadIdx.x;
  const int w = tid >> 5;        // wave id 0..7
  const int lane = tid & 31;     // wave32 lane
  const int n16 = lane & 15;     // N / M index inside 16-lane group
  const int half = lane >> 4;    // lane-group half (carries K/M high bit)

  v8f zero = {};

  // ---- stage emb[b] (8x128) in both layouts, coalesced global reads ----
#pragma unroll
  for (int i = 0; i < 4; ++i) {
    int idx = tid + i * 256;  // p*128 + j
    float v = emb[b * 1024 + idx];
    embA[idx] = v;
    int p = idx >> 7, j = idx & 127;
    embT[j * 8 + p] = v;
  }
  __syncthreads();

  v8f outacc = zero;  // out-tile accumulator across heads (wave w -> N-tile w)

  for (int h = 0; h < 8; ++h) {
    // ================= qT = Wq_h^T @ embT  (M=128 c, N=16(8 valid p), K=128)
    {
      v8f acc = zero;
      const int m = 16 * w + n16;  // c row
#pragma unroll 4
      for (int k0 = 0; k0 < 128; k0 += 4) {
        const int k = k0 + 2 * half;
        v2f a, bb;
        a.x = Wq[k * 1024 + m * 8 + h];
        a.y = Wq[(k + 1) * 1024 + m * 8 + h];
        bb.x = embT[k * 8 + (n16 & 7)];
        bb.y = embT[(k + 1) * 8 + (n16 & 7)];
        acc = wmma4(a, bb, acc);
      }
      if (n16 < 8) {
#pragma unroll
        for (int i = 0; i < 8; ++i) qT[(16 * w + i + 8 * half) * 8 + n16] = acc[i];
      }
    }
    // ================= kB = embA @ Wk_h    (M=16(8 valid p), N=128 d, K=128)
    {
      v8f acc = zero;
#pragma unroll 4
      for (int k0 = 0; k0 < 128; k0 += 4) {
        const int k = k0 + 2 * half;
        v2f a, bb;
        a.x = embA[(n16 & 7) * 128 + k];
        a.y = embA[(n16 & 7) * 128 + k + 1];
        bb.x = Wk[k * 1024 + (16 * w + n16) * 8 + h];
        bb.y = Wk[(k + 1) * 1024 + (16 * w + n16) * 8 + h];
        acc = wmma4(a, bb, acc);
      }
      if (half == 0) {  // rows m=i (p=0..7) valid
#pragma unroll
        for (int i = 0; i < 8; ++i) kB[i * 128 + 16 * w + n16] = acc[i];
      }
    }
    // ================= vT = Wv_h^T @ embT  (like qT)
    {
      v8f acc = zero;
      const int m = 16 * w + n16;
#pragma unroll 4
      for (int k0 = 0; k0 < 128; k0 += 4) {
        const int k = k0 + 2 * half;
        v2f a, bb;
        a.x = Wv[k * 1024 + m * 8 + h];
        a.y = Wv[(k + 1) * 1024 + m * 8 + h];
        bb.x = embT[k * 8 + (n16 & 7)];
        bb.y = embT[(k + 1) * 8 + (n16 & 7)];
        acc = wmma4(a, bb, acc);
      }
      if (n16 < 8) {
#pragma unroll
        for (int i = 0; i < 8; ++i) vT[(16 * w + i + 8 * half) * 8 + n16] = acc[i];
      }
    }
    __syncthreads();

    // ================= scores = qT @ kB  (wave w owns rows 16w..16w+15; K=8)
    v8f sc[8];
#pragma unroll
    for (int nt = 0; nt < 8; ++nt) sc[nt] = zero;
#pragma unroll
    for (int k0 = 0; k0 < 8; k0 += 4) {
      const int k = k0 + 2 * half;
      v2f a;
      a.x = qT[(16 * w + n16) * 8 + k];
      a.y = qT[(16 * w + n16) * 8 + k + 1];
#pragma unroll
      for (int nt = 0; nt < 8; ++nt) {
        v2f bb;
        bb.x = kB[k * 128 + nt * 16 + n16];
        bb.y = kB[(k + 1) * 128 + nt * 16 + n16];
        sc[nt] = wmma4(a, bb, sc[nt]);
      }
    }

    // ================= InstanceNorm stats over full 128x128 map
    float s = 0.f, ss = 0.f;
#pragma unroll
    for (int nt = 0; nt < 8; ++nt) {
#pragma unroll
      for (int i = 0; i < 8; ++i) {
        float v = sc[nt][i];
        s += v;
        ss += v * v;
      }
    }
#pragma unroll
    for (int off = 16; off > 0; off >>= 1) {
      s += __shfl_xor(s, off);
      ss += __shfl_xor(ss, off);
    }
    if (lane == 0) {
      redbuf[w] = s;
      redbuf[8 + w] = ss;
    }
    __syncthreads();
    float S = 0.f, SS = 0.f;
#pragma unroll
    for (int i = 0; i < 8; ++i) {
      S += redbuf[i];
      SS += redbuf[8 + i];
    }
    const float mean = S * (1.0f / 16384.0f);
    const float var = SS * (1.0f / 16384.0f) - mean * mean;
    const float rstd = rsqrtf(var + IN_EPS);

    // ================= normalize + row softmax
    // D-frag row m = i + 8*half lives entirely inside one 16-lane group,
    // so shfl_xor with masks 8,4,2,1 reduces exactly over that row.
#pragma unroll
    for (int i = 0; i < 8; ++i) {
      float mx = -3.402823466e38f;
#pragma unroll
      for (int nt = 0; nt < 8; ++nt) {
        float v = (sc[nt][i] - mean) * rstd;
        sc[nt][i] = v;
        mx = fmaxf(mx, v);
      }
#pragma unroll
      for (int off = 8; off > 0; off >>= 1) mx = fmaxf(mx, __shfl_xor(mx, off));
      float sum = 0.f;
#pragma unroll
      for (int nt = 0; nt < 8; ++nt) {
        float e = __expf(sc[nt][i] - mx);
        sc[nt][i] = e;
        sum += e;
      }
#pragma unroll
      for (int off = 8; off > 0; off >>= 1) sum += __shfl_xor(sum, off);
      const float inv = 1.0f / sum;
#pragma unroll
      for (int nt = 0; nt < 8; ++nt) sc[nt][i] *= inv;
    }

    // ================= ctx = attn @ vT  (K=128 in two staged halves of 64)
    v8f ctx = zero;
#pragma unroll
    for (int hb = 0; hb < 2; ++hb) {
      __syncthreads();  // attnL free (also fences prior-head out-GEMM)
#pragma unroll
      for (int nt = 0; nt < 4; ++nt) {
#pragma unroll
        for (int i = 0; i < 8; ++i)
          attnL[(16 * w + i + 8 * half) * 64 + nt * 16 + n16] = sc[hb * 4 + nt][i];
      }
      __syncthreads();
#pragma unroll 4
      for (int k0 = 0; k0 < 64; k0 += 4) {
        const int k = k0 + 2 * half;
        const int kg = hb * 64 + k;
        v2f a, bb;
        a.x = attnL[(16 * w + n16) * 64 + k];
        a.y = attnL[(16 * w + n16) * 64 + k + 1];
        bb.x = vT[kg * 8 + (n16 & 7)];
        bb.y = vT[(kg + 1) * 8 + (n16 & 7)];
        ctx = wmma4(a, bb, ctx);
      }
    }

    // ================= stage ctx^T [p][c] for this head
    __syncthreads();
    if (n16 < 8) {
#pragma unroll
      for (int i = 0; i < 8; ++i) ctxT[n16 * 128 + 16 * w + i + 8 * half] = ctx[i];
    }
    __syncthreads();

    // ================= out += ctx^T @ Wout_h  (M=16(8 p), N=128, K=128 c)
#pragma unroll 4
    for (int k0 = 0; k0 < 128; k0 += 4) {
      const int k = k0 + 2 * half;
      v2f a, bb;
      a.x = ctxT[(n16 & 7) * 128 + k];
      a.y = ctxT[(n16 & 7) * 128 + k + 1];
      bb.x = Wout[(k * 8 + h) * 128 + 16 * w + n16];
      bb.y = Wout[((k + 1) * 8 + h) * 128 + 16 * w + n16];
      outacc = wmma4(a, bb, outacc);
    }
  }  // heads

  // ---- final store with '(b d h w)(p1 p2 p3) -> (b_, d p1 h p2 w p3)' rows
  if (half == 0) {  // valid token rows p = i
    const int b_ = b >> 5, dd = (b >> 4) & 1, hs = (b >> 2) & 3, wsp = b & 3;
#pragma unroll
    for (int i = 0; i < 8; ++i) {
      const int p1 = i >> 2, p2 = (i >> 1) & 1, p3 = i & 1;
      const int r = b_ * 256 + dd * 128 + p1 * 64 + hs * 16 + p2 * 8 + wsp * 2 + p3;
      out[r * 128 + 16 * w + n16] = outacc[i];
    }
  }
}

extern "C" void kernel_launch(void* const* d_in, const int* in_sizes, int n_in,
                              void* d_out, int out_size, void* d_ws, size_t ws_size,
                              hipStream_t stream) {
  const float* emb = (const float*)d_in[0];
  const float* Wq = (const float*)d_in[1];
  const float* Wk = (const float*)d_in[2];
  const float* Wv = (const float*)d_in[3];
  const float* Wout = (const float*)d_in[4];
  float* out = (float*)d_out;
  (void)in_sizes; (void)n_in; (void)out_size; (void)d_ws; (void)ws_size;
  fused_cross_attn_local<<<dim3(1024), dim3(256), 0, stream>>>(emb, Wq, Wk, Wv,
                                                               Wout, out);
}